// SpecAugmentTime_51307679318730
// MI455X (gfx1250) — compile-verified
//
#include <hip/hip_runtime.h>

// SpecAugment time-masking: out[b,c,t] = (t in any mask interval of batch b) ? 0 : x[b,c,t]
// Shapes: x (16, 512, 4000) fp32; widths/starts (16, 2) int32.
//
// Pure streaming op: 262 MB traffic, ~0 flops/byte -> HBM-bound (~11.3 us at 23.3 TB/s).
// Strategy: b128 vector loads/stores with non-temporal hints, scalar loads for the
// wave-uniform per-batch mask bounds, wave32-friendly 256-thread blocks.

#define B_DIM      16
#define C_DIM      512
#define T_DIM      4000
#define NMASK      2
#define TPB        256
#define V4_PER_ROW (T_DIM / 4)   // 1000 float4 per (b,c) row; 16000 B row stride -> 16B aligned

// Native clang vector type: required by __builtin_nontemporal_load/store
// (HIP's float4 is a class and is rejected).
typedef float v4f __attribute__((ext_vector_type(4)));

__global__ __launch_bounds__(TPB)
void specaug_time_kernel(const float* __restrict__ x,
                         const int*   __restrict__ widths,
                         const int*   __restrict__ starts,
                         float*       __restrict__ out)
{
    // blockIdx.y indexes the (b, c) row; b is wave-uniform -> scalar loads for mask params.
    const int row = blockIdx.y;                 // 0 .. B*C-1
    const int b   = row >> 9;                   // row / C_DIM (C_DIM == 512)

    const int v4 = blockIdx.x * TPB + threadIdx.x;   // float4 index within the row
    if (v4 >= V4_PER_ROW) return;

    // Per-batch mask intervals [lo, hi): hi = start + width + 1. Uniform -> s_load path.
    const int lo0 = starts[b * NMASK + 0];
    const int hi0 = lo0 + widths[b * NMASK + 0] + 1;
    const int lo1 = starts[b * NMASK + 1];
    const int hi1 = lo1 + widths[b * NMASK + 1] + 1;

    const int idx = row * T_DIM + v4 * 4;       // < 32.77M, fits int32
    const int t0  = v4 * 4;

    // Streaming 128-bit non-temporal load (th:TH_LOAD_NT on gfx1250).
    const v4f v = __builtin_nontemporal_load((const v4f*)(x + idx));

    v4f r;
#pragma unroll
    for (int k = 0; k < 4; ++k) {
        const int t = t0 + k;
        const bool m = (t >= lo0 && t < hi0) || (t >= lo1 && t < hi1);
        r[k] = m ? 0.0f : v[k];
    }

    // Streaming 128-bit non-temporal store (th:TH_STORE_NT on gfx1250).
    __builtin_nontemporal_store(r, (v4f*)(out + idx));
}

extern "C" void kernel_launch(void* const* d_in, const int* in_sizes, int n_in,
                              void* d_out, int out_size, void* d_ws, size_t ws_size,
                              hipStream_t stream)
{
    (void)in_sizes; (void)n_in; (void)out_size; (void)d_ws; (void)ws_size;

    const float* x      = (const float*)d_in[0];
    const int*   widths = (const int*)d_in[1];
    const int*   starts = (const int*)d_in[2];
    float*       out    = (float*)d_out;

    dim3 grid((V4_PER_ROW + TPB - 1) / TPB,   // 4 blocks cover one 4000-float row
              B_DIM * C_DIM);                 // 8192 rows
    specaug_time_kernel<<<grid, TPB, 0, stream>>>(x, widths, starts, out);
}